// ParcorCoefficientsToLinearPredictiveCoefficients_21878563405828
// MI455X (gfx1250) — compile-verified
//
#include <hip/hip_runtime.h>
#include <stdint.h>

// PARCOR -> LPC (Levinson step-up), 2,097,152 rows x 25 f32. Bandwidth-bound:
// ~420 MB traffic vs ~1.16 GFLOP -> optimize data movement, not math.
// CDNA5 path: async global<->LDS b128 copies (ASYNCcnt) for coalescing of the
// 100-byte (unaligned) rows; per-row recurrence fully unrolled in registers.
// Note: INST_OFFSET on async-to/from-LDS ops applies to BOTH the LDS and the
// global address, so one (lds,voff) VGPR pair serves all chunks via offset:N.

#define NC 25                      // LPC_ORDER + 1
#define RPB 256                    // rows per block (= threads per block)
#define TILE_ELEMS (NC * RPB)      // 6400 floats = 25600 bytes LDS (16B-aligned tile)

__global__ __launch_bounds__(RPB)
void parcor2lpc_kernel(const float* __restrict__ k, float* __restrict__ out, int nrows)
{
    __shared__ __align__(16) float tile[TILE_ELEMS];

    const int tid = threadIdx.x;
    const long long rowBase  = (long long)blockIdx.x * RPB;
    const long long elemBase = rowBase * NC;

    float a[NC];

    if (rowBase + RPB <= (long long)nrows) {
        // ---------- fast path: full tile, async-staged through LDS ----------
        const uint32_t ldsBase = (uint32_t)(uintptr_t)(&tile[0]);
        const uint32_t off16   = (uint32_t)tid * 16u;   // b128 lane offset
        const uint32_t off4    = (uint32_t)tid * 4u;    // b32 tail lane offset
        const uint32_t lds16   = ldsBase + off16;
        const uint32_t lds4    = ldsBase + off4;

        const float* gsrc = k + elemBase;               // 25600*blockIdx bytes: 16B-aligned

        // 6 x b128 (bytes [0,24576)) + 1 x b32 (bytes [24576,25600)), fully coalesced
#define ALD128(J) asm volatile(                                              \
        "global_load_async_to_lds_b128 %0, %1, %2 offset:" #J " th:TH_LOAD_NT" \
        :: "v"(lds16), "v"(off16), "s"(gsrc) : "memory")
        ALD128(0);
        ALD128(4096);
        ALD128(8192);
        ALD128(12288);
        ALD128(16384);
        ALD128(20480);
#undef ALD128
        asm volatile("global_load_async_to_lds_b32 %0, %1, %2 offset:24576 th:TH_LOAD_NT"
                     :: "v"(lds4), "v"(off4), "s"(gsrc) : "memory");

        asm volatile("s_wait_asynccnt 0" ::: "memory");
        __syncthreads();

        // row for this thread; stride 25 dwords is coprime to 64 banks -> no conflicts
#pragma unroll
        for (int i = 0; i < NC; ++i)
            a[i] = tile[tid * NC + i];

        // ---------- step-up recurrence, fully unrolled in registers ----------
#pragma unroll
        for (int m = 2; m <= NC - 1; ++m) {
            const float km = a[m];       // a[m] == k[m] still untouched at step m
#pragma unroll
            for (int i = 1; 2 * i < m; ++i) {
                const float lo = a[i];
                const float hi = a[m - i];
                a[i]     = lo + km * hi;
                a[m - i] = hi + km * lo;
            }
            if ((m & 1) == 0) {
                const float mid = a[m >> 1];
                a[m >> 1] = mid + km * mid;
            }
        }

        // write back into own row (thread-private region: no barrier needed here)
#pragma unroll
        for (int i = 0; i < NC; ++i)
            tile[tid * NC + i] = a[i];
        __syncthreads();                 // all rows in LDS before async store

        // coalesced LDS -> global
        float* gdst = out + elemBase;
#define AST128(J) asm volatile(                                               \
        "global_store_async_from_lds_b128 %0, %1, %2 offset:" #J " th:TH_STORE_NT" \
        :: "v"(off16), "v"(lds16), "s"(gdst) : "memory")
        AST128(0);
        AST128(4096);
        AST128(8192);
        AST128(12288);
        AST128(16384);
        AST128(20480);
#undef AST128
        asm volatile("global_store_async_from_lds_b32 %0, %1, %2 offset:24576 th:TH_STORE_NT"
                     :: "v"(off4), "v"(lds4), "s"(gdst) : "memory");

        asm volatile("s_wait_asynccnt 0" ::: "memory");
    } else {
        // ---------- tail path: direct per-thread row (no async staging) ----------
        const long long row = rowBase + tid;
        if (row < (long long)nrows) {
            const float* src = k + row * NC;
#pragma unroll
            for (int i = 0; i < NC; ++i) a[i] = src[i];

#pragma unroll
            for (int m = 2; m <= NC - 1; ++m) {
                const float km = a[m];
#pragma unroll
                for (int i = 1; 2 * i < m; ++i) {
                    const float lo = a[i];
                    const float hi = a[m - i];
                    a[i]     = lo + km * hi;
                    a[m - i] = hi + km * lo;
                }
                if ((m & 1) == 0) {
                    const float mid = a[m >> 1];
                    a[m >> 1] = mid + km * mid;
                }
            }

            float* dst = out + row * NC;
#pragma unroll
            for (int i = 0; i < NC; ++i) dst[i] = a[i];
        }
    }
}

extern "C" void kernel_launch(void* const* d_in, const int* in_sizes, int n_in,
                              void* d_out, int out_size, void* d_ws, size_t ws_size,
                              hipStream_t stream) {
    (void)n_in; (void)d_ws; (void)ws_size; (void)out_size;
    const float* k = (const float*)d_in[0];
    float* out = (float*)d_out;

    const int nrows  = in_sizes[0] / NC;               // 2,097,152 for the reference
    const int blocks = (nrows + RPB - 1) / RPB;        // 8192 full blocks

    parcor2lpc_kernel<<<blocks, RPB, 0, stream>>>(k, out, nrows);
}